// MultiHeadEMA_66932770341590
// MI455X (gfx1250) — compile-verified
//
#include <hip/hip_runtime.h>
#include <cstdint>

// MultiHeadEMA as a chunked linear-recurrence scan (no FFT).
// out[b,s,d] = silu( sum_n w[d,n]*h_n[b,s,d] + omega[d]*x[b,s,d] )
//   h_n[s] = q[d,n]*h_n[s-1] + x[s],  q = 1 - sigmoid(delta)*sigmoid(alpha),
//   w = sigmoid(delta)*beta*gamma*sqrt(1/N)
//
// Phase 0: per-channel parameter precompute (q, w) -> ws
// Phase 1: per (b, 32-ch block, chunk) local scan -> chunk end states (part)
// Phase 2: inter-chunk prefix scan over 64 chunks (carry decay q^128)
// Phase 3: local scan again with corrected entry state, + residual + silu -> out
//
// Phases 1/3 stream x via async global->LDS copies (CDNA5 ASYNCcnt path),
// double-buffered 16-row groups, 4x b128 ops per group. Output stores are
// non-temporal so the 134MB x working set stays resident in the 192MB L2
// between phase 1 and phase 3 (second x read becomes an L2 hit).

#define B_  4
#define S_  8192
#define D_  1024
#define N_  2
#define C_  64            // chunks
#define T_  128           // chunk length = S_/C_
#define DBLKS_ (D_/32)    // 32 channel-blocks
#define WAVES_PER_BLK 8
#define SCALE_ 0.70710678118654752f

// ---------------- CDNA5 async global->LDS helpers ----------------

typedef int v4i_async __attribute__((vector_size(16)));

__device__ __forceinline__ void async_copy_b128(const float* gp, float* lp) {
#if defined(__gfx1250__) && __has_builtin(__builtin_amdgcn_global_load_async_to_lds_b128)
  __builtin_amdgcn_global_load_async_to_lds_b128(
      (__attribute__((address_space(1))) v4i_async*)(uintptr_t)gp,
      (__attribute__((address_space(3))) v4i_async*)(uint32_t)(uintptr_t)lp,
      0, 0);
#else
  // VDST = 32-bit LDS byte address, VADDR = 64-bit global address
  uint32_t l = (uint32_t)(uintptr_t)lp;
  uint64_t g = (uint64_t)(uintptr_t)gp;
  asm volatile("global_load_async_to_lds_b128 %0, %1, off"
               :: "v"(l), "v"(g) : "memory");
#endif
}

template <int CNT>
__device__ __forceinline__ void wait_async() {
#if defined(__gfx1250__) && __has_builtin(__builtin_amdgcn_s_wait_asynccnt)
  __builtin_amdgcn_s_wait_asynccnt(CNT);
#else
  asm volatile("s_wait_asynccnt %0" :: "i"(CNT) : "memory");
#endif
  asm volatile("" ::: "memory");   // compiler ordering fence for LDS reads
}

__device__ __forceinline__ void store_nt(float v, float* p) {
  __builtin_nontemporal_store(v, p);   // TH=NT: don't pollute L2 with streamed output
}

// ---------------- Phase 0: parameters ----------------

__global__ void __launch_bounds__(256)
ema_params(const float* __restrict__ delta, const float* __restrict__ alpha,
           const float* __restrict__ beta,  const float* __restrict__ gamma,
           float* __restrict__ wsq, float* __restrict__ wsw)
{
  const int t = blockIdx.x * blockDim.x + threadIdx.x;   // t = d*N + n
  if (t >= D_ * N_) return;
  const float p = 1.0f / (1.0f + __expf(-delta[t]));
  const float a = 1.0f / (1.0f + __expf(-alpha[t]));
  wsq[t] = 1.0f - p * a;
  wsw[t] = p * beta[t] * gamma[t] * SCALE_;
}

// ---------------- Phase 2: inter-chunk prefix scan ----------------

__global__ void __launch_bounds__(256)
ema_chunk_scan(float* __restrict__ part, const float* __restrict__ wsq)
{
  const int t = blockIdx.x * blockDim.x + threadIdx.x;   // B_*D_*N_ threads
  const int b = t / (D_ * N_);
  const int r = t % (D_ * N_);
  const int n = r / D_;
  const int d = r % D_;
  const float q = wsq[d * N_ + n];
  float qT = q;
  #pragma unroll
  for (int i = 0; i < 7; ++i) qT *= qT;                  // q^128 = q^T_
  float carry = 0.0f;                                    // state entering chunk 0
  for (int c = 0; c < C_; ++c) {
    const size_t idx = (((size_t)b * C_ + c) * N_ + n) * D_ + d;
    const float v = part[idx];                           // local end-state of chunk c
    part[idx] = carry;                                   // -> state entering chunk c
    carry = fmaf(qT, carry, v);
  }
}

// ---------------- Phases 1 & 3: streaming local scan ----------------

template <int FINAL>
__global__ void __launch_bounds__(256)
ema_pass(const float* __restrict__ x,   const float* __restrict__ wsq,
         const float* __restrict__ wsw, const float* __restrict__ omega,
         float* __restrict__ part,      float* __restrict__ out)
{
  __shared__ __align__(16) float smem[WAVES_PER_BLK][2][16][32];

  const int wv   = threadIdx.x >> 5;
  const int lane = threadIdx.x & 31;
  const int W    = blockIdx.x * WAVES_PER_BLK + wv;      // 8192 waves total
  const int c    = W % C_;
  const int dblk = (W / C_) & (DBLKS_ - 1);
  const int b    = W / (C_ * DBLKS_);
  const int d    = (dblk << 5) + lane;

  const float q0 = wsq[d * N_ + 0];
  const float q1 = wsq[d * N_ + 1];
  float h0 = 0.0f, h1 = 0.0f, w0 = 0.0f, w1 = 0.0f, om = 0.0f;
  if (FINAL) {
    w0 = wsw[d * N_ + 0];
    w1 = wsw[d * N_ + 1];
    om = omega[d];
    h0 = part[(((size_t)b * C_ + c) * N_ + 0) * D_ + d]; // entry states
    h1 = part[(((size_t)b * C_ + c) * N_ + 1) * D_ + d];
  }

  const size_t gbase = ((size_t)b * S_ + (size_t)c * T_) * (size_t)D_
                     + (size_t)(dblk << 5);
  const int lrow = lane >> 3;         // 0..3  : row within 4-row packet
  const int lcol = (lane & 7) << 2;   // 0..28 : float column (16B granules)

  // Issue one 16-row group (4 async b128 ops; each moves 4 rows x 128B)
  auto issue = [&](int g, int buf) {
    const float* gpg = x + gbase + (size_t)(g * 16 + lrow) * D_ + lcol;
    float*       lpg = &smem[wv][buf][lrow][lcol];
    #pragma unroll
    for (int i = 0; i < 4; ++i)
      async_copy_b128(gpg + (size_t)(i * 4) * D_, lpg + i * 4 * 32);
  };

  issue(0, 0);
  #pragma unroll 1
  for (int g = 0; g < 8; ++g) {       // 8 groups x 16 rows = T_ = 128
    const int buf = g & 1;
    if (g < 7) { issue(g + 1, buf ^ 1); wait_async<4>(); }
    else       { wait_async<0>(); }
    #pragma unroll
    for (int r = 0; r < 16; ++r) {
      const float xv = smem[wv][buf][r][lane];
      h0 = fmaf(q0, h0, xv);
      h1 = fmaf(q1, h1, xv);
      if (FINAL) {
        const float y  = fmaf(w0, h0, fmaf(w1, h1, om * xv));
        const float sg = 1.0f / (1.0f + __expf(-y));     // silu
        store_nt(y * sg, &out[gbase + (size_t)(g * 16 + r) * D_ + lane]);
      }
    }
  }

  if (!FINAL) {
    store_nt(h0, &part[(((size_t)b * C_ + c) * N_ + 0) * D_ + d]);
    store_nt(h1, &part[(((size_t)b * C_ + c) * N_ + 1) * D_ + d]);
  }
}

// ---------------- launch ----------------

extern "C" void kernel_launch(void* const* d_in, const int* in_sizes, int n_in,
                              void* d_out, int out_size, void* d_ws, size_t ws_size,
                              hipStream_t stream) {
  const float* x     = (const float*)d_in[0];
  const float* delta = (const float*)d_in[1];
  const float* alpha = (const float*)d_in[2];
  const float* beta  = (const float*)d_in[3];
  const float* gamma = (const float*)d_in[4];
  const float* omega = (const float*)d_in[5];
  float* out = (float*)d_out;

  // workspace layout (floats): q[D*N] | w[D*N] | part[B*C*N*D]  (~2.1 MB)
  float* wsq  = (float*)d_ws;
  float* wsw  = wsq + D_ * N_;
  float* part = wsw + D_ * N_;

  const int passBlocks = (B_ * DBLKS_ * C_) / WAVES_PER_BLK;   // 1024

  ema_params<<<(D_ * N_ + 255) / 256, 256, 0, stream>>>(delta, alpha, beta, gamma, wsq, wsw);
  ema_pass<0><<<passBlocks, 256, 0, stream>>>(x, wsq, wsw, omega, part, out);
  ema_chunk_scan<<<(B_ * D_ * N_) / 256, 256, 0, stream>>>(part, wsq);
  ema_pass<1><<<passBlocks, 256, 0, stream>>>(x, wsq, wsw, omega, part, out);
}